// ImprovedCalciumVQVAE_30030411333975
// MI455X (gfx1250) — compile-verified
//
#include <hip/hip_runtime.h>
#include <hip/hip_bf16.h>

typedef float v2f __attribute__((ext_vector_type(2)));
typedef float v8f __attribute__((ext_vector_type(8)));

// Explicit global-address-space pointers so the hot loops lower to
// global_load_b32 (not flat_load_b32, which ties up the LDS path and DScnt).
typedef __attribute__((address_space(1))) const float* gcf;
typedef __attribute__((address_space(1))) float*       gf;

struct Taps { int kidx[8]; int off[8]; };

#define NT 4   // position tiles per wave (N = 16*NT)

// ---------------------------------------------------------------------------
// Generic implicit-GEMM 1D conv via V_WMMA_F32_16X16X4_F32.
// y[b, o, m*out_stride + out_phase] = relu?( bias[o] + res[...] +
//     sum_tap sum_i w[o, i, kidx[tap]] * x[b, i, m*in_stride + off[tap]] )
// One wave computes a 16 (channels) x 64 (positions) macro-tile: the weight
// (A) fragment is loaded once per K-step and reused by NT=4 WMMAs, and the 4
// independent accumulator chains let the scheduler overlap loads with WMMA.
// K-dim is traversed tap-major so the WMMA K sub-index maps directly to the
// channel index (no division in the hot loop); the bounds check per tap/tile
// is loop-invariant and applied with a branchless cndmask on clamped loads.
// All indexing is integer offsets off addrspace(1) bases.
// ---------------------------------------------------------------------------
__global__ __launch_bounds__(32)
void conv1d_wmma_kernel(const float* xp, const float* wp, const float* biasp,
                        const float* resp, float* yp,
                        int I, int O, int Lin, int Lout_full,
                        int in_stride, int out_stride, int out_phase,
                        int Kw, int KT, Taps taps, int relu)
{
    gcf x    = (gcf)xp;
    gcf w    = (gcf)wp;
    gcf bias = (gcf)biasp;
    gcf res  = (gcf)resp;
    gf  y    = (gf)yp;

    const int lane = threadIdx.x;
    const int n    = lane & 15;              // column (time) / A row (channel)
    const int half = lane >> 4;
    const int kc   = half * 2;               // K sub-index base for this half-wave
    const int ob   = blockIdx.x * 16;        // output-channel tile base
    const int pb   = blockIdx.y * (16 * NT); // position macro-tile base
    const int b    = blockIdx.z;

    const int orow = ob + n;
    const long long xbase = (long long)b * I * Lin;
    const long long wrow_base = (long long)orow * I * Kw;

    v8f acc[NT];
    #pragma unroll
    for (int j = 0; j < NT; ++j) acc[j] = (v8f){0.f,0.f,0.f,0.f,0.f,0.f,0.f,0.f};

    for (int tap = 0; tap < KT; ++tap) {
        bool va[NT];
        long long xo[NT];
        #pragma unroll
        for (int j = 0; j < NT; ++j) {
            const int pj = (pb + j * 16 + n) * in_stride + taps.off[tap];
            va[j] = (pj >= 0) && (pj < Lin);
            xo[j] = xbase + (va[j] ? pj : 0) + (long long)kc * Lin;
        }
        long long wo = wrow_base + taps.kidx[tap] + (long long)kc * Kw;
        for (int i0 = 0; i0 < I; i0 += 4) {
            v2f a;
            a.x = w[wo];
            a.y = w[wo + Kw];
            #pragma unroll
            for (int j = 0; j < NT; ++j) {
                const float x0 = x[xo[j]];
                const float x1 = x[xo[j] + Lin];
                v2f bf;
                bf.x = va[j] ? x0 : 0.f;
                bf.y = va[j] ? x1 : 0.f;
                acc[j] = __builtin_amdgcn_wmma_f32_16x16x4_f32(false, a, false, bf,
                                                               (short)0, acc[j],
                                                               false, false);
                xo[j] += 4 * (long long)Lin;
            }
            wo += 4 * (long long)Kw;
        }
    }

    const long long ybase = (long long)b * O * Lout_full;
    const bool has_bias = (biasp != nullptr);
    const bool has_res  = (resp != nullptr);
    #pragma unroll
    for (int j = 0; j < NT; ++j) {
        const int outp = (pb + j * 16 + n) * out_stride + out_phase;
        #pragma unroll
        for (int r = 0; r < 8; ++r) {
            const int o = ob + half * 8 + r;
            float v = acc[j][r];
            if (has_bias) v += bias[o];
            const long long yi = ybase + (long long)o * Lout_full + outp;
            if (has_res) v += res[yi];
            if (relu) v = fmaxf(v, 0.f);
            y[yi] = v;
        }
    }
}

// ---------------------------------------------------------------------------
// GroupNorm (+ ReLU). One block per (batch, group).
// ---------------------------------------------------------------------------
__global__ __launch_bounds__(256)
void groupnorm_relu_kernel(const float* __restrict__ x, const float* __restrict__ g,
                           const float* __restrict__ bta, float* __restrict__ y,
                           int C, int L, int groups, int relu)
{
    const int b = blockIdx.y, grp = blockIdx.x;
    const int cpg = C / groups;
    const long long base = ((long long)b * C + (long long)grp * cpg) * L;
    const int nelem = cpg * L;
    float s = 0.f, s2 = 0.f;
    for (int i = threadIdx.x; i < nelem; i += blockDim.x) {
        const float v = x[base + i];
        s += v; s2 += v * v;
    }
    __shared__ float sh[256], sh2[256];
    sh[threadIdx.x] = s; sh2[threadIdx.x] = s2;
    __syncthreads();
    for (int o = 128; o > 0; o >>= 1) {
        if ((int)threadIdx.x < o) {
            sh[threadIdx.x]  += sh[threadIdx.x + o];
            sh2[threadIdx.x] += sh2[threadIdx.x + o];
        }
        __syncthreads();
    }
    const float mean = sh[0] / (float)nelem;
    const float var  = sh2[0] / (float)nelem - mean * mean;
    const float rstd = rsqrtf(var + 1e-5f);
    for (int i = threadIdx.x; i < nelem; i += blockDim.x) {
        const int cl = grp * cpg + i / L;
        float v = (x[base + i] - mean) * rstd * g[cl] + bta[cl];
        if (relu) v = fmaxf(v, 0.f);
        y[base + i] = v;
    }
}

// ---------------------------------------------------------------------------
// Fused vector-quantization: one wave per 16 time positions.
// scores = cb @ z via WMMA, argmin of (||c||^2 - 2*score), write quantized,
// indices, per-block loss partial (deterministic), histogram (exact float).
// ---------------------------------------------------------------------------
__global__ __launch_bounds__(32)
void vq_kernel(const float* zp_, const float* cbp_, const float* cbnp_,
               float* qoutp_, int* idxout, float* blockloss,
               float* hist, int L, int D, int K)
{
    gcf z    = (gcf)zp_;
    gcf cb   = (gcf)cbp_;
    gcf cbn  = (gcf)cbnp_;
    gf  qout = (gf)qoutp_;

    __shared__ float zt[128 * 16];
    __shared__ int sidx[16];
    const int lane = threadIdx.x;
    const int tb = blockIdx.x * 16;
    const int b  = blockIdx.y;
    const long long zbase = (long long)b * D * L;

    for (int i = lane; i < D * 16; i += 32) {
        const int d = i >> 4, t = i & 15;
        zt[i] = z[zbase + (long long)d * L + tb + t];
    }
    __syncthreads();

    const int n = lane & 15, half = lane >> 4, kc = half * 2;
    float znorm = 0.f;
    if (lane < 16) {
        for (int d = 0; d < D; ++d) { const float v = zt[d * 16 + n]; znorm += v * v; }
    }

    float minv = 3.4e38f; int mini = 0;
    for (int ct = 0; ct < K / 16; ++ct) {
        v8f c = {0.f, 0.f, 0.f, 0.f, 0.f, 0.f, 0.f, 0.f};
        long long cbo = (long long)(ct * 16 + n) * D + kc;
        int zo = kc * 16 + n;
        for (int kk0 = 0; kk0 < D; kk0 += 4) {
            v2f a, bf;
            a.x  = cb[cbo];
            a.y  = cb[cbo + 1];
            bf.x = zt[zo];
            bf.y = zt[zo + 16];
            c = __builtin_amdgcn_wmma_f32_16x16x4_f32(false, a, false, bf,
                                                      (short)0, c, false, false);
            cbo += 4;
            zo  += 64;
        }
        #pragma unroll
        for (int r = 0; r < 8; ++r) {
            const int code = ct * 16 + half * 8 + r;
            const float d2 = cbn[code] - 2.0f * c[r];
            if (d2 < minv) { minv = d2; mini = code; }
        }
    }
    // combine lane n (codes M=0..7 of each tile) with lane n+16 (M=8..15)
    const float ov = __shfl_xor(minv, 16, 32);
    const int   oi = __shfl_xor(mini, 16, 32);
    if (ov < minv || (ov == minv && oi < mini)) { minv = ov; mini = oi; }

    float lv = 0.f;
    if (lane < 16) {
        idxout[b * L + tb + n] = mini;
        sidx[n] = mini;
        atomicAdd(&hist[mini], 1.0f);   // integer-valued: exact, order-independent
        lv = znorm + minv;              // ||z||^2 + ||c||^2 - 2 z.c = ||q - z||^2
    }
    for (int off = 16; off >= 1; off >>= 1) lv += __shfl_down(lv, off, 32);
    if (lane == 0) blockloss[blockIdx.y * gridDim.x + blockIdx.x] = lv;
    __syncthreads();

    for (int i = lane; i < D * 16; i += 32) {
        const int d = i >> 4, t = i & 15;
        qout[zbase + (long long)d * L + tb + t] = cb[(long long)sidx[t] * D + d];
    }
}

// ---------------------------------------------------------------------------
// Small helper kernels
// ---------------------------------------------------------------------------
__global__ void fill_zero_kernel(float* p, long long nelem) {
    long long i = (long long)blockIdx.x * blockDim.x + threadIdx.x;
    const long long stride = (long long)gridDim.x * blockDim.x;
    for (; i < nelem; i += stride) p[i] = 0.f;
}

__global__ void scatter_onehot_kernel(const int* __restrict__ idx,
                                      float* __restrict__ enc, int n, int K) {
    const int i = blockIdx.x * blockDim.x + threadIdx.x;
    if (i < n) enc[(long long)i * K + idx[i]] = 1.0f;
}

__global__ void cbnorm_kernel(const float* __restrict__ cb, float* __restrict__ cbn,
                              int K, int D) {
    const int c = blockIdx.x * blockDim.x + threadIdx.x;
    if (c < K) {
        float s = 0.f;
        for (int d = 0; d < D; ++d) { const float v = cb[(long long)c * D + d]; s += v * v; }
        cbn[c] = s;
    }
}

__global__ __launch_bounds__(1024)
void finalize_kernel(const float* __restrict__ hist, const float* __restrict__ blockloss,
                     int nblocks, float* __restrict__ out_loss, float* __restrict__ out_perp)
{
    __shared__ float sh[1024];
    const int t = threadIdx.x;
    const float avg = hist[t] * (1.0f / 32768.0f);
    sh[t] = avg * logf(avg + 1e-10f);
    __syncthreads();
    for (int o = 512; o > 0; o >>= 1) { if (t < o) sh[t] += sh[t + o]; __syncthreads(); }
    const float ent = sh[0];
    __syncthreads();
    float ls = 0.f;
    for (int i = t; i < nblocks; i += 1024) ls += blockloss[i];
    sh[t] = ls;
    __syncthreads();
    for (int o = 512; o > 0; o >>= 1) { if (t < o) sh[t] += sh[t + o]; __syncthreads(); }
    if (t == 0) {
        *out_perp = expf(-ent);
        *out_loss = 0.25f * sh[0] / (32768.0f * 128.0f);
    }
}

__global__ void pool_kernel(const float* __restrict__ q, float* __restrict__ zp,
                            int BD, int L) {
    const int i = blockIdx.x * blockDim.x + threadIdx.x;
    if (i < BD) {
        const long long base = (long long)i * L;
        float s = 0.f;
        for (int t = 0; t < L; ++t) s += q[base + t];
        zp[i] = s / (float)L;
    }
}

__global__ void linear_kernel(const float* __restrict__ x, const float* __restrict__ w,
                              const float* __restrict__ b, float* __restrict__ y,
                              int Bn, int In, int On, int relu) {
    const int i = blockIdx.x * blockDim.x + threadIdx.x;
    if (i < Bn * On) {
        const int bb = i / On, o = i % On;
        float s = b[o];
        for (int d = 0; d < In; ++d) s += x[bb * In + d] * w[o * In + d];
        if (relu) s = fmaxf(s, 0.f);
        y[i] = s;
    }
}

// ---------------------------------------------------------------------------
// Host launch
// ---------------------------------------------------------------------------
extern "C" void kernel_launch(void* const* d_in, const int* in_sizes, int n_in,
                              void* d_out_v, int out_size, void* d_ws, size_t ws_size,
                              hipStream_t stream)
{
    const float* x    = (const float*)d_in[0];
    const float* e1w  = (const float*)d_in[1];  const float* e1b = (const float*)d_in[2];
    const float* e2w  = (const float*)d_in[3];  const float* e2b = (const float*)d_in[4];
    const float* e3w  = (const float*)d_in[5];  const float* e3b = (const float*)d_in[6];
    const float* rg1g[2] = {(const float*)d_in[7],  (const float*)d_in[13]};
    const float* rg1b[2] = {(const float*)d_in[8],  (const float*)d_in[14]};
    const float* rc1w[2] = {(const float*)d_in[9],  (const float*)d_in[15]};
    const float* rg2g[2] = {(const float*)d_in[10], (const float*)d_in[16]};
    const float* rg2b[2] = {(const float*)d_in[11], (const float*)d_in[17]};
    const float* rc2w[2] = {(const float*)d_in[12], (const float*)d_in[18]};
    const float* pvw  = (const float*)d_in[19]; const float* pvb = (const float*)d_in[20];
    const float* cb   = (const float*)d_in[21];
    const float* d1w  = (const float*)d_in[22]; const float* d1b = (const float*)d_in[23];
    const float* d2w  = (const float*)d_in[24]; const float* d2b = (const float*)d_in[25];
    const float* d3w  = (const float*)d_in[26]; const float* d3b = (const float*)d_in[27];
    const float* bh1w = (const float*)d_in[28]; const float* bh1b = (const float*)d_in[29];
    const float* bh2w = (const float*)d_in[30]; const float* bh2b = (const float*)d_in[31];
    const float* bh3w = (const float*)d_in[32]; const float* bh3b = (const float*)d_in[33];

    float* out = (float*)d_out_v;
    float* ws  = (float*)d_ws;

    // output layout (flat, in tuple return order)
    float* out_loss = out;                       // [1]
    float* xrec     = out + 1;                   // [32*128*4096]
    float* out_perp = out + 16777217LL;          // [1]
    float* qout     = out + 16777218LL;          // [32*128*1024]
    float* enc      = out + 20971522LL;          // [32768*1024]
    float* beh      = out + 54525954LL;          // [32*4]

    // workspace layout (floats) with reuse
    float* h1   = ws;                 // 16.78M
    float* h2   = ws + 16777216LL;    // 16.78M
    float* w2b  = ws + 33554432LL;    // 16.78M  e3 out / residual accumulator
    float* w3b  = ws;                 // gn1 output (h1 dead after e2)
    float* w4b  = ws + 16777216LL;    // 4.19M   (h2 dead after e3)
    float* zb   = ws + 20971520LL;    // 4.19M   pre-VQ z
    float* dec1 = ws;                 // 33.55M  (encoder temps dead after VQ)
    float* dec2 = ws + 33554432LL;    // 33.55M  (w2b dead after pv)
    float* cbn       = ws + 67108864LL;   // 1024
    float* hist      = cbn + 1024;        // 1024
    float* blockloss = hist + 1024;       // 2048
    int*   idxarr    = (int*)(blockloss + 2048); // 32768
    float* zpool     = (float*)(idxarr + 32768); // 4096
    float* bt1       = zpool + 4096;      // 4096
    float* bt2       = bt1 + 4096;        // 2048

    // tap tables
    Taps t7{}; for (int k = 0; k < 7; ++k) { t7.kidx[k] = k; t7.off[k] = k - 3; }
    Taps t5{}; for (int k = 0; k < 5; ++k) { t5.kidx[k] = k; t5.off[k] = k - 2; }
    Taps t3{}; for (int k = 0; k < 3; ++k) { t3.kidx[k] = k; t3.off[k] = k - 1; }
    Taps t1{}; t1.kidx[0] = 0; t1.off[0] = 0;
    Taps d1e{}; d1e.kidx[0] = 1; d1e.off[0] = 0;
    Taps d1o{}; d1o.kidx[0] = 0; d1o.off[0] = 0; d1o.kidx[1] = 2; d1o.off[1] = 1;
    Taps d2e{}; d2e.kidx[0] = 0; d2e.off[0] = -1; d2e.kidx[1] = 2; d2e.off[1] = 0;
               d2e.kidx[2] = 4; d2e.off[2] = 1;
    Taps d2o{}; d2o.kidx[0] = 1; d2o.off[0] = 0; d2o.kidx[1] = 3; d2o.off[1] = 1;

    auto conv = [&](const float* xin, const float* wgt, const float* bias,
                    const float* res, float* yout, int I, int O, int Lin,
                    int Lout_full, int Lm, int in_stride, int out_stride,
                    int out_phase, int Kw, int KT, const Taps& tp, int relu) {
        dim3 grid(O / 16, Lm / (16 * NT), 32);
        hipLaunchKernelGGL(conv1d_wmma_kernel, grid, dim3(32), 0, stream,
                           xin, wgt, bias, res, yout, I, O, Lin, Lout_full,
                           in_stride, out_stride, out_phase, Kw, KT, tp, relu);
    };
    auto gn = [&](const float* xin, const float* g, const float* bt, float* yout,
                  int C, int L) {
        hipLaunchKernelGGL(groupnorm_relu_kernel, dim3(8, 32), dim3(256), 0, stream,
                           xin, g, bt, yout, C, L, 8, 1);
    };

    // ---- encoder ----
    conv(x,  e1w, e1b, nullptr, h1,  128, 128, 4096, 4096, 4096, 1, 1, 0, 7, 7, t7, 1);
    conv(h1, e2w, e2b, nullptr, h2,  128, 256, 4096, 2048, 2048, 2, 1, 0, 5, 5, t5, 1);
    conv(h2, e3w, e3b, nullptr, w2b, 256, 512, 2048, 1024, 1024, 2, 1, 0, 3, 3, t3, 0);

    // ---- residual blocks ----
    for (int r = 0; r < 2; ++r) {
        gn(w2b, rg1g[r], rg1b[r], w3b, 512, 1024);
        conv(w3b, rc1w[r], nullptr, nullptr, w4b, 512, 128, 1024, 1024, 1024, 1, 1, 0, 3, 3, t3, 0);
        gn(w4b, rg2g[r], rg2b[r], w4b, 128, 1024);
        conv(w4b, rc2w[r], nullptr, w2b, w2b, 128, 512, 1024, 1024, 1024, 1, 1, 0, 1, 1, t1, 0);
    }

    // ---- pre-VQ 1x1 conv ----
    conv(w2b, pvw, pvb, nullptr, zb, 512, 128, 1024, 1024, 1024, 1, 1, 0, 1, 1, t1, 0);

    // ---- vector quantization ----
    hipLaunchKernelGGL(cbnorm_kernel, dim3(4), dim3(256), 0, stream, cb, cbn, 1024, 128);
    hipLaunchKernelGGL(fill_zero_kernel, dim3(4), dim3(256), 0, stream, hist, 1024LL);
    hipLaunchKernelGGL(vq_kernel, dim3(64, 32), dim3(32), 0, stream,
                       zb, cb, cbn, qout, idxarr, blockloss, hist, 1024, 128, 1024);
    hipLaunchKernelGGL(fill_zero_kernel, dim3(4096), dim3(256), 0, stream, enc, 33554432LL);
    hipLaunchKernelGGL(scatter_onehot_kernel, dim3(128), dim3(256), 0, stream,
                       idxarr, enc, 32768, 1024);
    hipLaunchKernelGGL(finalize_kernel, dim3(1), dim3(1024), 0, stream,
                       hist, blockloss, 2048, out_loss, out_perp);

    // ---- decoder (ConvTranspose as phase convs) ----
    conv(qout, d1w, d1b, nullptr, dec1, 128, 512, 1024, 2048, 1024, 1, 2, 0, 3, 1, d1e, 1);
    conv(qout, d1w, d1b, nullptr, dec1, 128, 512, 1024, 2048, 1024, 1, 2, 1, 3, 2, d1o, 1);
    conv(dec1, d2w, d2b, nullptr, dec2, 512, 256, 2048, 4096, 2048, 1, 2, 0, 5, 3, d2e, 1);
    conv(dec1, d2w, d2b, nullptr, dec2, 512, 256, 2048, 4096, 2048, 1, 2, 1, 5, 2, d2o, 1);
    conv(dec2, d3w, d3b, nullptr, xrec, 256, 128, 4096, 4096, 4096, 1, 1, 0, 7, 7, t7, 0);

    // ---- behavior head ----
    hipLaunchKernelGGL(pool_kernel, dim3(16), dim3(256), 0, stream, qout, zpool, 4096, 1024);
    hipLaunchKernelGGL(linear_kernel, dim3(16), dim3(256), 0, stream,
                       zpool, bh1w, bh1b, bt1, 32, 128, 128, 1);
    hipLaunchKernelGGL(linear_kernel, dim3(8), dim3(256), 0, stream,
                       bt1, bh2w, bh2b, bt2, 32, 128, 64, 1);
    hipLaunchKernelGGL(linear_kernel, dim3(1), dim3(128), 0, stream,
                       bt2, bh3w, bh3b, beh, 32, 64, 4, 0);
}